// MyModel_87454124082093
// MI455X (gfx1250) — compile-verified
//
#include <hip/hip_runtime.h>
#include <hip/hip_bf16.h>

// y[i] = x[i] > 0 ? x[i]*x[i] : x[i]
// Strictly HBM-bound on MI455X: 512 MB of traffic / 23.3 TB/s ~= 23 us floor.
// Hot path = unguarded full-tile kernel: 4 x global_load_b128 (NT) in flight
// per lane, branchless v_cndmask select, 4 x global_store_b128 (NT).
// Bounds handling is hoisted into separately-launched remainder kernels so the
// main kernel has zero exec-mask manipulation and clean load clustering.

typedef float v4f __attribute__((ext_vector_type(4)));

#define BLOCK 256
#define ITEMS 4  // v4f per thread => 64 B loaded + 64 B stored per thread

__device__ __forceinline__ v4f sqrelu4(v4f v) {
  v4f r;
  r.x = (v.x > 0.0f) ? v.x * v.x : v.x;
  r.y = (v.y > 0.0f) ? v.y * v.y : v.y;
  r.z = (v.z > 0.0f) ? v.z * v.z : v.z;
  r.w = (v.w > 0.0f) ? v.w * v.w : v.w;
  return r;
}

// Full tiles only: no bounds checks, no exec-mask churn.
__global__ __launch_bounds__(BLOCK) void sqrelu_full_kernel(
    const v4f* __restrict__ x, v4f* __restrict__ y) {
  unsigned int base = blockIdx.x * (BLOCK * ITEMS) + threadIdx.x;

  v4f v[ITEMS];
#pragma unroll
  for (int k = 0; k < ITEMS; ++k) {
    v[k] = __builtin_nontemporal_load(&x[base + (unsigned int)k * BLOCK]);
  }
#pragma unroll
  for (int k = 0; k < ITEMS; ++k) {
    __builtin_nontemporal_store(sqrelu4(v[k]), &y[base + (unsigned int)k * BLOCK]);
  }
}

// Remainder v4f elements (start4..n4): one guarded v4f per thread.
__global__ __launch_bounds__(BLOCK) void sqrelu_rem_kernel(
    const v4f* __restrict__ x, v4f* __restrict__ y,
    unsigned int start4, unsigned int n4) {
  unsigned int i = start4 + blockIdx.x * BLOCK + threadIdx.x;
  if (i < n4) {
    v4f v = __builtin_nontemporal_load(&x[i]);
    __builtin_nontemporal_store(sqrelu4(v), &y[i]);
  }
}

// Scalar tail (n % 4 != 0; never runs for N = 2^26).
__global__ __launch_bounds__(64) void sqrelu_tail_kernel(
    const float* __restrict__ x, float* __restrict__ y,
    unsigned int start, unsigned int n) {
  unsigned int i = start + blockIdx.x * 64u + threadIdx.x;
  if (i < n) {
    float v = x[i];
    y[i] = (v > 0.0f) ? v * v : v;
  }
}

extern "C" void kernel_launch(void* const* d_in, const int* in_sizes, int n_in,
                              void* d_out, int out_size, void* d_ws, size_t ws_size,
                              hipStream_t stream) {
  (void)n_in; (void)d_ws; (void)ws_size; (void)out_size;
  const float* x = (const float*)d_in[0];
  float* y = (float*)d_out;
  unsigned int n = (unsigned int)in_sizes[0];

  unsigned int n4 = n / 4u;
  const unsigned int per_block = BLOCK * ITEMS;  // v4f per full block

  unsigned int full_blocks = n4 / per_block;
  if (full_blocks > 0) {
    sqrelu_full_kernel<<<full_blocks, BLOCK, 0, stream>>>(
        (const v4f*)x, (v4f*)y);
  }

  unsigned int done4 = full_blocks * per_block;
  unsigned int rem4 = n4 - done4;
  if (rem4 > 0) {
    sqrelu_rem_kernel<<<(rem4 + BLOCK - 1u) / BLOCK, BLOCK, 0, stream>>>(
        (const v4f*)x, (v4f*)y, done4, n4);
  }

  unsigned int rem = n - n4 * 4u;
  if (rem > 0) {
    sqrelu_tail_kernel<<<(rem + 63u) / 64u, 64, 0, stream>>>(x, y, n4 * 4u, n);
  }
}